// GATLayer_68195490726428
// MI455X (gfx1250) — compile-verified
//
#include <hip/hip_runtime.h>
#include <hip/hip_bf16.h>
#include <math.h>

typedef __attribute__((ext_vector_type(2))) float v2f;
typedef __attribute__((ext_vector_type(8))) float v8f;

#define HEADS 4
#define F_OUT 16
#define F_TOT 64            // HEADS * F_OUT
#define K_DIM 128
#define NEG_SLOPE 0.2f

// ---------------------------------------------------------------------------
// Kernel 1: xp = x @ W^T  via V_WMMA_F32_16X16X4_F32
//   x  : [N, 128] row-major
//   W  : [64, 128] row-major  (B[k][n] = W[n][k])
//   xp : [N, 64]  row-major
// One wave computes one 16x16 output tile. Block = 128 threads = 4 waves,
// wave w handles column tile w (cols 16w..16w+15). Grid = N/16 row tiles.
// ---------------------------------------------------------------------------
__global__ __launch_bounds__(128) void gat_gemm_wmma(
    const float* __restrict__ x, const float* __restrict__ W,
    float* __restrict__ xp, int N) {
  const int wave = threadIdx.x >> 5;
  const int lane = threadIdx.x & 31;
  const int row0 = blockIdx.x * 16;
  const int col0 = wave * 16;

  // A fragment: lane m = lane&15 holds A[m][kbase], A[m][kbase+1]
  // B fragment: lane n = lane&15 holds B[kbase][n], B[kbase+1][n] = W[n][kbase..]
  const int mn = lane & 15;
  const int kb = (lane >> 4) * 2;

  const float* arow = x + (size_t)(row0 + mn) * K_DIM + kb;
  const float* brow = W + (size_t)(col0 + mn) * K_DIM + kb;

  v8f c = {};
#pragma unroll
  for (int k = 0; k < K_DIM; k += 4) {
    v2f a = *(const v2f*)(arow + k);
    v2f b = *(const v2f*)(brow + k);
    // 8 args: (neg_a, A, neg_b, B, c_mod, C, reuse_a, reuse_b)
    c = __builtin_amdgcn_wmma_f32_16x16x4_f32(
        false, a, false, b, (short)0, c, false, false);
  }

  // C/D layout: VGPR r -> row = r + (lane>>4)*8, col = lane&15
  const int rbase = row0 + (lane >> 4) * 8;
  const int n = col0 + (lane & 15);
#pragma unroll
  for (int r = 0; r < 8; ++r) {
    xp[(size_t)(rbase + r) * F_TOT + n] = c[r];
  }
}

// ---------------------------------------------------------------------------
// Kernel 2: per-(node, head) attention logits
//   a_src[n,h] = dot(xp[n, h*16 : h*16+16], att_src[h])
//   a_dst[n,h] = dot(xp[n, h*16 : h*16+16], att_dst[h])
// ---------------------------------------------------------------------------
__global__ void gat_logits(const float* __restrict__ xp,
                           const float* __restrict__ att_src,
                           const float* __restrict__ att_dst,
                           float* __restrict__ a_src, float* __restrict__ a_dst,
                           int N) {
  int t = blockIdx.x * blockDim.x + threadIdx.x;   // t = node*4 + h
  if (t >= N * HEADS) return;
  const int h = t & 3;
  const int node = t >> 2;
  const float* v = xp + (size_t)node * F_TOT + h * F_OUT;
  const float* as = att_src + h * F_OUT;
  const float* ad = att_dst + h * F_OUT;
  float s = 0.f, d = 0.f;
#pragma unroll
  for (int f = 0; f < F_OUT; ++f) {
    float xv = v[f];
    s += xv * as[f];
    d += xv * ad[f];
  }
  a_src[t] = s;
  a_dst[t] = d;
}

// ---------------------------------------------------------------------------
// Kernel 3: init m = -inf, denom = 0, out[n,f] = bias[f]
// ---------------------------------------------------------------------------
__global__ void gat_init(float* __restrict__ mbuf, float* __restrict__ denom,
                         float* __restrict__ out, const float* __restrict__ bias,
                         int N) {
  int t = blockIdx.x * blockDim.x + threadIdx.x;
  if (t < N * HEADS) {
    ((unsigned int*)mbuf)[t] = 0xFF800000u;  // -inf
    denom[t] = 0.f;
  }
  long long tt = (long long)blockIdx.x * blockDim.x + threadIdx.x;
  if (tt < (long long)N * F_TOT) {
    out[tt] = bias[tt & (F_TOT - 1)];
  }
}

// deterministic float atomic max via sign-split int/uint atomics
__device__ __forceinline__ void atomicMaxF(float* addr, float v) {
  if (v >= 0.f)
    atomicMax((int*)addr, __float_as_int(v));
  else
    atomicMin((unsigned int*)addr, __float_as_uint(v));
}

__device__ __forceinline__ void edge_endpoints(const long long* __restrict__ ei,
                                               long long eid, int E,
                                               int& s, int& d) {
  if (eid < (long long)E) {
    s = (int)ei[eid];
    d = (int)ei[(long long)E + eid];
  } else {              // self loop
    s = d = (int)(eid - E);
  }
}

// ---------------------------------------------------------------------------
// Kernel 4: segment max of leakyrelu(a_src[s]+a_dst[d]) per (dst, head)
// one thread per (edge, head)
// ---------------------------------------------------------------------------
__global__ void gat_segmax(const long long* __restrict__ ei,
                           const float* __restrict__ a_src,
                           const float* __restrict__ a_dst,
                           float* __restrict__ mbuf, int E, int N) {
  long long t = (long long)blockIdx.x * blockDim.x + threadIdx.x;
  long long total = (long long)(E + N) * HEADS;
  if (t >= total) return;
  const int h = (int)(t & 3);
  const long long eid = t >> 2;
  int s, d;
  edge_endpoints(ei, eid, E, s, d);
  float e = a_src[s * HEADS + h] + a_dst[d * HEADS + h];
  e = (e > 0.f) ? e : e * NEG_SLOPE;
  atomicMaxF(&mbuf[d * HEADS + h], e);
}

// ---------------------------------------------------------------------------
// Kernel 5: denom[d,h] += exp(e - m[d,h]); one thread per (edge, head)
// ---------------------------------------------------------------------------
__global__ void gat_segsum(const long long* __restrict__ ei,
                           const float* __restrict__ a_src,
                           const float* __restrict__ a_dst,
                           const float* __restrict__ mbuf,
                           float* __restrict__ denom, int E, int N) {
  long long t = (long long)blockIdx.x * blockDim.x + threadIdx.x;
  long long total = (long long)(E + N) * HEADS;
  if (t >= total) return;
  const int h = (int)(t & 3);
  const long long eid = t >> 2;
  int s, d;
  edge_endpoints(ei, eid, E, s, d);
  float e = a_src[s * HEADS + h] + a_dst[d * HEADS + h];
  e = (e > 0.f) ? e : e * NEG_SLOPE;
  float m = mbuf[d * HEADS + h];
  if (!(m > -3.0e38f)) m = 0.f;   // mirror reference isfinite guard
  atomicAdd(&denom[d * HEADS + h], expf(e - m));
}

// ---------------------------------------------------------------------------
// Kernel 6: out[d, f] += alpha[e, f>>4] * xp[s, f]; one thread per (edge, f)
// 64 consecutive threads share an edge -> edge_index loads broadcast from L0.
// ---------------------------------------------------------------------------
__global__ void gat_aggregate(const long long* __restrict__ ei,
                              const float* __restrict__ xp,
                              const float* __restrict__ a_src,
                              const float* __restrict__ a_dst,
                              const float* __restrict__ mbuf,
                              const float* __restrict__ denom,
                              float* __restrict__ out, int E, int N) {
  long long t = (long long)blockIdx.x * blockDim.x + threadIdx.x;
  long long total = (long long)(E + N) * F_TOT;
  if (t >= total) return;
  const int f = (int)(t & (F_TOT - 1));
  const long long eid = t >> 6;
  const int h = f >> 4;
  int s, d;
  edge_endpoints(ei, eid, E, s, d);
  float e = a_src[s * HEADS + h] + a_dst[d * HEADS + h];
  e = (e > 0.f) ? e : e * NEG_SLOPE;
  float m = mbuf[d * HEADS + h];
  if (!(m > -3.0e38f)) m = 0.f;
  float alpha = expf(e - m) / (denom[d * HEADS + h] + 1e-16f);
  atomicAdd(&out[(size_t)d * F_TOT + f], alpha * xp[(size_t)s * F_TOT + f]);
}

// ---------------------------------------------------------------------------
extern "C" void kernel_launch(void* const* d_in, const int* in_sizes, int n_in,
                              void* d_out, int out_size, void* d_ws, size_t ws_size,
                              hipStream_t stream) {
  const float* x          = (const float*)d_in[0];       // [N,128]
  const long long* ei     = (const long long*)d_in[1];   // [2,E] int64
  const float* W          = (const float*)d_in[2];       // [64,128]
  const float* att_src    = (const float*)d_in[3];       // [4,16]
  const float* att_dst    = (const float*)d_in[4];       // [4,16]
  const float* bias       = (const float*)d_in[5];       // [64]
  float* out              = (float*)d_out;               // [N,64]

  const int N = in_sizes[0] / K_DIM;      // 100000
  const int E = in_sizes[1] / 2;          // 1600000

  // workspace layout
  float* xp    = (float*)d_ws;                       // N*64
  float* a_src = xp    + (size_t)N * F_TOT;          // N*4
  float* a_dst = a_src + (size_t)N * HEADS;          // N*4
  float* mbuf  = a_dst + (size_t)N * HEADS;          // N*4
  float* denom = mbuf  + (size_t)N * HEADS;          // N*4

  // 1) GEMM via WMMA: grid = N/16 row tiles, 4 waves = 4 col tiles per block
  gat_gemm_wmma<<<(N + 15) / 16, 128, 0, stream>>>(x, W, xp, N);

  // 2) per-node logits
  {
    int total = N * HEADS;
    gat_logits<<<(total + 255) / 256, 256, 0, stream>>>(xp, att_src, att_dst,
                                                        a_src, a_dst, N);
  }

  // 3) init m / denom / out(bias)
  {
    long long total = (long long)N * F_TOT;
    int blocks = (int)((total + 255) / 256);
    gat_init<<<blocks, 256, 0, stream>>>(mbuf, denom, out, bias, N);
  }

  // 4) segment max
  {
    long long total = (long long)(E + N) * HEADS;
    int blocks = (int)((total + 255) / 256);
    gat_segmax<<<blocks, 256, 0, stream>>>(ei, a_src, a_dst, mbuf, E, N);
  }

  // 5) segment sum of exp
  {
    long long total = (long long)(E + N) * HEADS;
    int blocks = (int)((total + 255) / 256);
    gat_segsum<<<blocks, 256, 0, stream>>>(ei, a_src, a_dst, mbuf, denom, E, N);
  }

  // 6) weighted aggregation
  {
    long long total = (long long)(E + N) * F_TOT;
    int blocks = (int)((total + 255) / 256);
    gat_aggregate<<<blocks, 256, 0, stream>>>(ei, xp, a_src, a_dst, mbuf, denom,
                                              out, E, N);
  }
}